// TGNNodeClassifier_56049323213477
// MI455X (gfx1250) — compile-verified
//
#include <hip/hip_runtime.h>
#include <math.h>

// ---------------------------------------------------------------------------
// TGN node classifier for gfx1250 (MI455X).
// Dense GEMMs use V_WMMA_F32_16X16X4_F32 (exact fp32; problem is
// memory/atomic-bound at ~2.5 GB / 23.3 TB/s, so no reason to drop precision).
// ---------------------------------------------------------------------------

#define N_NODES   100000
#define N_EVENTS  200000
#define DMSG      64
#define DTIME     32
#define DFEAT     96          // [msg(64) | time_enc(32)] -- zero memory cols dropped
#define DMEM      128
#define DGNN      128
#define NCLS      64
#define N_SEDGES  1600000
#define N_TEDGES  (N_SEDGES + N_NODES)   // + self loops

typedef float v2f __attribute__((ext_vector_type(2)));
typedef float v8f __attribute__((ext_vector_type(8)));

__device__ __forceinline__ unsigned f2key(float f) {
  unsigned b = __float_as_uint(f);
  return (b & 0x80000000u) ? ~b : (b | 0x80000000u);
}
__device__ __forceinline__ float key2f(unsigned k) {
  unsigned b = (k & 0x80000000u) ? (k & 0x7fffffffu) : ~k;
  return __uint_as_float(b);
}

// --- 1) per-node last event: lexicographic (time, eid) max, packed u64 ------
__global__ void k_scatter_last(const int* __restrict__ src, const int* __restrict__ dst,
                               const int* __restrict__ t,
                               unsigned long long* __restrict__ packed) {
  int i = blockIdx.x * blockDim.x + threadIdx.x;
  if (i >= 2 * N_EVENTS) return;
  int e    = (i < N_EVENTS) ? i : i - N_EVENTS;
  int node = (i < N_EVENTS) ? src[e] : dst[e];
  unsigned long long v = ((unsigned long long)(unsigned)(t[e] + 1) << 32) | (unsigned)i;
  atomicMax(&packed[node], v);
}

// --- 2) feat[n] = [msg[ev], cos(t*w+b)]; one wave32 per node ----------------
__global__ void k_build_feat(const unsigned long long* __restrict__ packed,
                             const int* __restrict__ t,
                             const float* __restrict__ msg,
                             const float* __restrict__ time_w,
                             const float* __restrict__ time_b,
                             float* __restrict__ feat) {
  int wave = (blockIdx.x * blockDim.x + threadIdx.x) >> 5;
  int lane = threadIdx.x & 31;
  if (wave >= N_NODES) return;
  float* frow = feat + (size_t)wave * DFEAT;
  unsigned long long p = packed[wave];
  if (p == 0ULL) {                      // node never got a message
    frow[lane] = 0.f; frow[lane + 32] = 0.f; frow[lane + 64] = 0.f;
    return;
  }
  int sel = (int)(unsigned)(p & 0xffffffffULL);
  int ev  = (sel < N_EVENTS) ? sel : sel - N_EVENTS;
  const float* mrow = msg + (size_t)ev * DMSG;
  frow[lane]      = mrow[lane];
  frow[lane + 32] = mrow[lane + 32];
  float tr = (float)t[ev];
  frow[lane + 64] = cosf(tr * time_w[lane] + time_b[lane]);
}

// --- generic fp32 WMMA GEMM: C[M x 16*colTiles] = op(A)[M x K] * W(:,woff:)^T
// One wave per 16x16 output tile.  A-fragment (16x4 fp32, ISA layout):
//   lanes 0-15 -> M=lane, v0={K+0,K+1}; lanes 16-31 -> v0={K+2,K+3}.
// B mirrors A (rows of the 4x16 tile); C/D: vgpr i = row i (+8 for hi lanes).
// abias != null: a = relu(a + abias[k]).  obias != null: += obias[col].
__global__ void k_wmma_gemm(const float* __restrict__ A, int lda,
                            const float* __restrict__ W, int ldw, int woff,
                            const float* __restrict__ abias,
                            const float* __restrict__ obias,
                            float* __restrict__ C, int ldc,
                            int K, int M, int colTiles) {
  int wave = (blockIdx.x * blockDim.x + threadIdx.x) >> 5;
  int lane = threadIdx.x & 31;
  int rowTile = wave / colTiles;
  int colTile = wave - rowTile * colTiles;
  int row0 = rowTile * 16;
  if (row0 >= M) return;                 // wave-uniform; EXEC stays all-ones
  int col0 = colTile * 16;
  int half = lane >> 4;
  int l15  = lane & 15;
  const float* Ap = A + (size_t)(row0 + l15) * lda;
  const float* Wp = W + (size_t)(col0 + l15) * ldw + woff;
  v8f acc = {};
  if (abias) {
    for (int k0 = 0; k0 < K; k0 += 4) {
      int kk = k0 + half * 2;
      v2f a  = *(const v2f*)(Ap + kk);
      v2f ab = *(const v2f*)(abias + kk);
      a.x = fmaxf(a.x + ab.x, 0.f);
      a.y = fmaxf(a.y + ab.y, 0.f);
      v2f b  = *(const v2f*)(Wp + kk);
      acc = __builtin_amdgcn_wmma_f32_16x16x4_f32(false, a, false, b,
                                                  (short)0, acc, false, false);
    }
  } else {
    for (int k0 = 0; k0 < K; k0 += 4) {
      int kk = k0 + half * 2;
      v2f a = *(const v2f*)(Ap + kk);
      v2f b = *(const v2f*)(Wp + kk);
      acc = __builtin_amdgcn_wmma_f32_16x16x4_f32(false, a, false, b,
                                                  (short)0, acc, false, false);
    }
  }
  float ob = obias ? obias[col0 + l15] : 0.f;
#pragma unroll
  for (int i = 0; i < 8; ++i) {
    int r = row0 + i + half * 8;
    C[(size_t)r * ldc + col0 + l15] = acc[i] + ob;
  }
}

// --- 4) GRU gates (memory == 0 => gh = b_hh only), masked by has_msg --------
__global__ void k_gru(const float* __restrict__ gi,
                      const unsigned long long* __restrict__ packed,
                      const float* __restrict__ b_ih, const float* __restrict__ b_hh,
                      float* __restrict__ mem) {
  int tid = blockIdx.x * blockDim.x + threadIdx.x;
  if (tid >= N_NODES * DMEM) return;
  int n = tid >> 7;
  int j = tid & 127;
  const float* g = gi + (size_t)n * 384;
  float gr = g[j]       + b_ih[j]       + b_hh[j];
  float gz = g[j + 128] + b_ih[j + 128] + b_hh[j + 128];
  float gn = g[j + 256] + b_ih[j + 256];
  float hn = b_hh[j + 256];
  float r  = 1.f / (1.f + __expf(-gr));
  float z  = 1.f / (1.f + __expf(-gz));
  float nn = tanhf(gn + r * hn);
  float h  = (1.f - z) * nn;             // + z * memory(=0)
  mem[tid] = (packed[n] != 0ULL) ? h : 0.f;
}

// --- 5) per-node attention dots; one wave per node --------------------------
__global__ void k_attn(const float* __restrict__ x,
                       const float* __restrict__ att_src, const float* __restrict__ att_dst,
                       float* __restrict__ a_s, float* __restrict__ a_d) {
  int wave = (blockIdx.x * blockDim.x + threadIdx.x) >> 5;
  int lane = threadIdx.x & 31;
  if (wave >= N_NODES) return;
  float4 xv = ((const float4*)(x + (size_t)wave * DGNN))[lane];
  float4 ws = ((const float4*)att_src)[lane];
  float4 wd = ((const float4*)att_dst)[lane];
  float ss = xv.x * ws.x + xv.y * ws.y + xv.z * ws.z + xv.w * ws.w;
  float sd = xv.x * wd.x + xv.y * wd.y + xv.z * wd.z + xv.w * wd.w;
  for (int off = 16; off > 0; off >>= 1) {
    ss += __shfl_xor(ss, off, 32);
    sd += __shfl_xor(sd, off, 32);
  }
  if (lane == 0) { a_s[wave] = ss; a_d[wave] = sd; }
}

__device__ __forceinline__ void edge_sd(const int* __restrict__ es,
                                        const int* __restrict__ ed,
                                        int i, int& s, int& d) {
  if (i < N_SEDGES) { s = es[i]; d = ed[i]; }
  else              { s = i - N_SEDGES; d = s; }   // self loop
}

// --- 6) softmax max (ordered-uint atomicMax) --------------------------------
__global__ void k_edge_max(const int* __restrict__ es, const int* __restrict__ ed,
                           const float* __restrict__ a_s, const float* __restrict__ a_d,
                           unsigned* __restrict__ mkey) {
  int i = blockIdx.x * blockDim.x + threadIdx.x;
  if (i >= N_TEDGES) return;
  int s, d; edge_sd(es, ed, i, s, d);
  float e = a_s[s] + a_d[d];
  e = (e > 0.f) ? e : 0.2f * e;          // leaky_relu(0.2)
  atomicMax(&mkey[d], f2key(e));
}

// --- 7) exp + denominator ---------------------------------------------------
__global__ void k_edge_exp(const int* __restrict__ es, const int* __restrict__ ed,
                           const float* __restrict__ a_s, const float* __restrict__ a_d,
                           const unsigned* __restrict__ mkey,
                           float* __restrict__ exbuf, float* __restrict__ den) {
  int i = blockIdx.x * blockDim.x + threadIdx.x;
  if (i >= N_TEDGES) return;
  int s, d; edge_sd(es, ed, i, s, d);
  float e = a_s[s] + a_d[d];
  e = (e > 0.f) ? e : 0.2f * e;
  float ex = __expf(e - key2f(mkey[d]));
  exbuf[i] = ex;
  atomicAdd(&den[d], ex);
}

// --- 8) weighted scatter: z[d] += alpha * x[s]; one wave per edge -----------
__global__ void k_edge_scatter(const int* __restrict__ es, const int* __restrict__ ed,
                               const float* __restrict__ exbuf, const float* __restrict__ den,
                               const float* __restrict__ x, float* __restrict__ zg) {
  int wave = (blockIdx.x * blockDim.x + threadIdx.x) >> 5;
  int lane = threadIdx.x & 31;
  if (wave >= N_TEDGES) return;
  int s, d; edge_sd(es, ed, wave, s, d);
  float alpha = exbuf[wave] / den[d];
  float4 xv = ((const float4*)(x + (size_t)s * DGNN))[lane];
  float* zr = zg + (size_t)d * DGNN + lane * 4;
  atomicAdd(zr + 0, alpha * xv.x);
  atomicAdd(zr + 1, alpha * xv.y);
  atomicAdd(zr + 2, alpha * xv.z);
  atomicAdd(zr + 3, alpha * xv.w);
}

// ---------------------------------------------------------------------------
extern "C" void kernel_launch(void* const* d_in, const int* in_sizes, int n_in,
                              void* d_out, int out_size, void* d_ws, size_t ws_size,
                              hipStream_t stream) {
  const int*   src      = (const int*)d_in[0];
  const int*   dst      = (const int*)d_in[1];
  const int*   t        = (const int*)d_in[2];
  const float* msg      = (const float*)d_in[3];
  const int*   sedge    = (const int*)d_in[4];
  const float* time_w   = (const float*)d_in[5];
  const float* time_b   = (const float*)d_in[6];
  const float* gru_w_ih = (const float*)d_in[7];
  const float* gru_b_ih = (const float*)d_in[9];
  const float* gru_b_hh = (const float*)d_in[10];
  const float* gat_w    = (const float*)d_in[11];
  const float* att_src  = (const float*)d_in[12];
  const float* att_dst  = (const float*)d_in[13];
  const float* gat_b    = (const float*)d_in[14];
  const float* cls_w    = (const float*)d_in[15];
  const float* cls_b    = (const float*)d_in[16];
  float* out = (float*)d_out;
  (void)in_sizes; (void)n_in; (void)out_size; (void)ws_size;

  // workspace layout (bytes); gi region is reused for x / z / ex after GRU
  char* ws = (char*)d_ws;
  unsigned long long* packed = (unsigned long long*)(ws + 0);              // 800 KB
  float*    feat  = (float*)(ws + 800000);                                 // 38.4 MB
  float*    memv  = (float*)(ws + 39200000);                               // 51.2 MB
  float*    gi    = (float*)(ws + 90400000);                               // 153.6 MB
  float*    x     = gi;                                                    // alias (after GRU)
  float*    zg    = (float*)(ws + 90400000 + 51200000);                    // alias
  float*    exbuf = (float*)(ws + 90400000 + 102400000);                   // alias
  float*    a_s   = (float*)(ws + 244000000);
  float*    a_d   = (float*)(ws + 244400000);
  unsigned* mkey  = (unsigned*)(ws + 244800000);
  float*    den   = (float*)(ws + 245200000);

  hipMemsetAsync(packed, 0, (size_t)N_NODES * 8, stream);
  hipMemsetAsync(mkey,   0, (size_t)N_NODES * 4, stream);
  hipMemsetAsync(den,    0, (size_t)N_NODES * 4, stream);

  k_scatter_last<<<(2 * N_EVENTS + 255) / 256, 256, 0, stream>>>(src, dst, t, packed);
  k_build_feat<<<(N_NODES * 32 + 255) / 256, 256, 0, stream>>>(packed, t, msg, time_w,
                                                               time_b, feat);
  // GEMM1: gi[N,384] = feat[N,96] @ gru_w_ih[:,256:352]^T   (6250 x 24 tiles)
  {
    int waves = (N_NODES / 16) * 24;
    k_wmma_gemm<<<waves * 32 / 256, 256, 0, stream>>>(feat, DFEAT, gru_w_ih, 352, 256,
        nullptr, nullptr, gi, 384, DFEAT, N_NODES, 24);
  }
  k_gru<<<(N_NODES * DMEM + 255) / 256, 256, 0, stream>>>(gi, packed, gru_b_ih,
                                                          gru_b_hh, memv);
  hipMemsetAsync(zg, 0, (size_t)N_NODES * DGNN * 4, stream);  // after gi is consumed
  // GEMM2: x[N,128] = mem @ gat_w^T                          (6250 x 8 tiles)
  {
    int waves = (N_NODES / 16) * 8;
    k_wmma_gemm<<<waves * 32 / 256, 256, 0, stream>>>(memv, DMEM, gat_w, DMEM, 0,
        nullptr, nullptr, x, DGNN, DMEM, N_NODES, 8);
  }
  k_attn<<<(N_NODES * 32 + 255) / 256, 256, 0, stream>>>(x, att_src, att_dst, a_s, a_d);

  const int* es = sedge;
  const int* ed = sedge + N_SEDGES;
  k_edge_max<<<(N_TEDGES + 255) / 256, 256, 0, stream>>>(es, ed, a_s, a_d, mkey);
  k_edge_exp<<<(N_TEDGES + 255) / 256, 256, 0, stream>>>(es, ed, a_s, a_d, mkey, exbuf, den);
  k_edge_scatter<<<(N_TEDGES * 32 + 255) / 256, 256, 0, stream>>>(es, ed, exbuf, den, x, zg);

  // GEMM3: out[N,64] = relu(zg + gat_b) @ cls_w^T + cls_b    (6250 x 4 tiles)
  {
    int waves = (N_NODES / 16) * 4;
    k_wmma_gemm<<<waves * 32 / 256, 256, 0, stream>>>(zg, DGNN, cls_w, DGNN, 0,
        gat_b, cls_b, out, NCLS, DGNN, N_NODES, 4);
  }
}